// OdeRollout_56538949484748
// MI455X (gfx1250) — compile-verified
//
#include <hip/hip_runtime.h>
#include <math.h>

// Problem constants (from reference)
#define TT   512
#define BB   256
#define DCH  48      // STATE_DIM + INPUT_DIM
#define HID  256
#define DT_C 0.01f

// LDS strides (padded to spread across the 64 banks, 8B-aligned)
#define LDX 50       // for 48-wide tiles
#define LDH 258      // for 256-wide tiles

typedef __attribute__((ext_vector_type(2))) float v2f;
typedef __attribute__((ext_vector_type(8))) float v8f;

// ---------------------------------------------------------------------------
// V_WMMA_F32_16X16X4_F32 wrapper: D = A(16x4,f32) * B(4x16,f32) + C(16x16,f32)
// ---------------------------------------------------------------------------
__device__ __forceinline__ v8f wmma4_f32(v2f a, v2f b, v8f c) {
  return __builtin_amdgcn_wmma_f32_16x16x4_f32(false, a, false, b, (short)0, c,
                                               false, false);
}

// Hardware tanh (CDNA5 TRANS op v_tanh_f32) — replaces the ~40-instruction
// branchy libm expansion that dominated the VALU count in round 1.
__device__ __forceinline__ float fast_tanh(float x) {
#if __has_builtin(__builtin_amdgcn_tanhf)
  return __builtin_amdgcn_tanhf(x);
#else
  return tanhf(x);
#endif
}

// A-fragment 16x4 fp32 from LDS, row-major, stride lda (floats).
// Element A[m][k]: lane = m + 16*(k/2), vgpr = k%2.
__device__ __forceinline__ v2f frag_a_lds(const float* A, int lda, int k0) {
  const int l = threadIdx.x & 31;
  const int m = l & 15;
  const int k = k0 + ((l >> 4) << 1);
  v2f a;
  a.x = A[m * lda + k];
  a.y = A[m * lda + k + 1];
  return a;
}

// B-fragment 4x16 fp32 from global W (row-major, stride ldw floats).
// Element B[k][n]: lane = n + 16*(k/2), vgpr = k%2.
__device__ __forceinline__ v2f frag_b_glb(const float* __restrict__ W, int ldw,
                                          int k0, int n0) {
  const int l = threadIdx.x & 31;
  const int n = n0 + (l & 15);
  const int k = k0 + ((l >> 4) << 1);
  v2f b;
  b.x = W[k * ldw + n];
  b.y = W[(k + 1) * ldw + n];
  return b;
}

// Store 16x16 C tile to LDS with bias + tanh.
// Element C[m][n]: vgpr v, m = v + 8*(lane/16), n = lane%16.
__device__ __forceinline__ void store_c_bias_tanh(float* H, int ldh, v8f c,
                                                  const float* __restrict__ bias,
                                                  int n0) {
  const int l = threadIdx.x & 31;
  const int n = n0 + (l & 15);
  const int mh = (l >> 4) * 8;
  const float bn = bias[n];
#pragma unroll
  for (int v = 0; v < 8; ++v) {
    H[(mh + v) * ldh + n] = fast_tanh(c[v] + bn);
  }
}

// Store raw 16x16 C tile into a partials buffer P[wave][16][16].
__device__ __forceinline__ void store_c_partial(float* P, v8f c) {
  const int l = threadIdx.x & 31;
  const int n = l & 15;
  const int mh = (l >> 4) * 8;
  const int wave = threadIdx.x >> 5;
#pragma unroll
  for (int v = 0; v < 8; ++v) {
    P[(wave * 16 + mh + v) * 16 + n] = c[v];
  }
}

// Layer-2 core: 16x16 tile of tanh(A(16x256) @ W(256x256) + bias), N-tile n0.
// Two independent accumulator chains (even/odd K blocks) so the XDL pipe can
// overlap two in-flight accumulations instead of one serialized RAW chain.
__device__ __forceinline__ void gemm256_tile_tanh(
    const float* Alds, const float* __restrict__ W,
    const float* __restrict__ bias, float* Hout, int n0) {
  v8f c0 = {};
  v8f c1 = {};
#pragma unroll 8
  for (int k0 = 0; k0 < HID; k0 += 8) {
    const v2f a0 = frag_a_lds(Alds, LDH, k0);
    const v2f b0 = frag_b_glb(W, HID, k0, n0);
    c0 = wmma4_f32(a0, b0, c0);
    const v2f a1 = frag_a_lds(Alds, LDH, k0 + 4);
    const v2f b1 = frag_b_glb(W, HID, k0 + 4, n0);
    c1 = wmma4_f32(a1, b1, c1);
  }
  const v8f c = c0 + c1;  // deterministic merge of the two chains
  store_c_bias_tanh(Hout, LDH, c, bias, n0);
}

// ---------------------------------------------------------------------------
// Kernel 1: x_dd_pred = MLP(inputs) for ALL (b,t) rows -> out[..., 32:48]
// One WG (512 thr / 16 waves) per 16 rows. 8192 WGs total.
// ---------------------------------------------------------------------------
__global__ __launch_bounds__(512) void mlp_all_kernel(
    const float* __restrict__ inp, const float* __restrict__ W1,
    const float* __restrict__ b1, const float* __restrict__ W2,
    const float* __restrict__ b2, const float* __restrict__ W3,
    const float* __restrict__ b3, float* __restrict__ out) {
  __shared__ float sX[16 * LDX];
  __shared__ float sH1[16 * LDH];
  __shared__ float sH2[16 * LDH];
  __shared__ float sP[16 * 16 * 16];  // layer-3 K-split partials

  const int tid = threadIdx.x;
  const int wave = tid >> 5;
  const long long R = (long long)blockIdx.x * 16;  // first row of tile

  // Stage X tile (16 x 48): z_ts is just the raw input row.
  for (int i = tid; i < 16 * DCH; i += 512) {
    const int m = i / DCH, c = i % DCH;
    sX[m * LDX + c] = inp[(R + m) * DCH + c];
  }
  __syncthreads();

  // Layer 1: H1 = tanh(X @ W1 + b1); wave w owns N-tile w. K = 48.
  {
    const int n0 = wave * 16;
    v8f c = {};
#pragma unroll
    for (int k0 = 0; k0 < DCH; k0 += 4) {
      const v2f a = frag_a_lds(sX, LDX, k0);
      const v2f b = frag_b_glb(W1, HID, k0, n0);
      c = wmma4_f32(a, b, c);
    }
    store_c_bias_tanh(sH1, LDH, c, b1, n0);
  }
  __syncthreads();

  // Layer 2: H2 = tanh(H1 @ W2 + b2); wave w owns N-tile w, K = 256.
  gemm256_tile_tanh(sH1, W2, b2, sH2, wave * 16);
  __syncthreads();

  // Layer 3: OUT = H2 @ W3 + b3 (N = 16 only). Split K=256 across 16 waves.
  {
    const int k0w = wave * 16;
    v8f c = {};
#pragma unroll
    for (int kk = 0; kk < 16; kk += 4) {
      const v2f a = frag_a_lds(sH2, LDH, k0w + kk);
      const v2f b = frag_b_glb(W3, 16, k0w + kk, 0);
      c = wmma4_f32(a, b, c);
    }
    store_c_partial(sP, c);
  }
  __syncthreads();

  // Deterministic reduce (fixed order) + write channels 32..47.
  if (tid < 256) {
    const int m = tid >> 4, j = tid & 15;
    float s = b3[j];
#pragma unroll
    for (int w = 0; w < 16; ++w) s += sP[(w * 16 + m) * 16 + j];
    out[(R + m) * DCH + 32 + j] = s;
  }
}

// ---------------------------------------------------------------------------
// Kernel 2: Euler rollout. Batch rows are independent -> 16 WGs, each owns
// 16 trajectories and runs all 511 steps locally (workgroup barriers only).
// Writes out[..., 0:32] for all t.
// ---------------------------------------------------------------------------
__global__ __launch_bounds__(512) void rollout_kernel(
    const float* __restrict__ inp, const float* __restrict__ W1,
    const float* __restrict__ b1, const float* __restrict__ W2,
    const float* __restrict__ b2, const float* __restrict__ W3,
    const float* __restrict__ b3, float* __restrict__ out) {
  __shared__ float sZ[16 * LDX];
  __shared__ float sH1[16 * LDH];
  __shared__ float sH2[16 * LDH];
  __shared__ float sP[16 * 16 * 16];
  __shared__ float sY[16 * 34];  // state y, padded stride 34

  const int tid = threadIdx.x;
  const int wave = tid >> 5;
  const long long b0 = (long long)blockIdx.x * 16;  // first batch row

  // Init: y = inputs[:,0,0:32]; also emit y_pred at t=0.
  {
    const int m = tid >> 5, c = tid & 31;  // 512 threads = 16 rows x 32 ch
    const float v = inp[(b0 + m) * (long long)(TT * DCH) + c];
    sY[m * 34 + c] = v;
    out[(b0 + m) * (long long)(TT * DCH) + c] = v;
  }
  __syncthreads();

  for (int t = 1; t < TT; ++t) {
    // Build z = [y, tau_t] (16 x 48).
    for (int i = tid; i < 16 * DCH; i += 512) {
      const int m = i / DCH, c = i % DCH;
      const float v =
          (c < 32) ? sY[m * 34 + c]
                   : inp[(b0 + m) * (long long)(TT * DCH) + (long long)t * DCH + c];
      sZ[m * LDX + c] = v;
    }
    __syncthreads();

    // Layer 1 (K = 48).
    {
      const int n0 = wave * 16;
      v8f c = {};
#pragma unroll
      for (int k0 = 0; k0 < DCH; k0 += 4) {
        const v2f a = frag_a_lds(sZ, LDX, k0);
        const v2f b = frag_b_glb(W1, HID, k0, n0);
        c = wmma4_f32(a, b, c);
      }
      store_c_bias_tanh(sH1, LDH, c, b1, n0);
    }
    __syncthreads();

    // Layer 2 (K = 256, dual accumulator chains).
    gemm256_tile_tanh(sH1, W2, b2, sH2, wave * 16);
    __syncthreads();

    // Layer 3 partials (K split across waves).
    {
      const int k0w = wave * 16;
      v8f c = {};
#pragma unroll
      for (int kk = 0; kk < 16; kk += 4) {
        const v2f a = frag_a_lds(sH2, LDH, k0w + kk);
        const v2f b = frag_b_glb(W3, 16, k0w + kk, 0);
        c = wmma4_f32(a, b, c);
      }
      store_c_partial(sP, c);
    }
    __syncthreads();

    // Deterministic reduce + Euler update + store y_pred[:, t, :].
    if (tid < 256) {
      const int m = tid >> 4, j = tid & 15;
      float xdd = b3[j];
#pragma unroll
      for (int w = 0; w < 16; ++w) xdd += sP[(w * 16 + m) * 16 + j];
      const float ylo = sY[m * 34 + j];
      const float yhi = sY[m * 34 + 16 + j];
      const float nlo = ylo + DT_C * yhi;   // x += dt * x_dot
      const float nhi = yhi + DT_C * xdd;   // x_dot += dt * x_ddot
      sY[m * 34 + j] = nlo;
      sY[m * 34 + 16 + j] = nhi;
      const long long o =
          (b0 + m) * (long long)(TT * DCH) + (long long)t * DCH;
      out[o + j] = nlo;
      out[o + 16 + j] = nhi;
    }
    __syncthreads();
  }
}

// ---------------------------------------------------------------------------
extern "C" void kernel_launch(void* const* d_in, const int* in_sizes, int n_in,
                              void* d_out, int out_size, void* d_ws,
                              size_t ws_size, hipStream_t stream) {
  const float* inp = (const float*)d_in[0];
  const float* W1 = (const float*)d_in[1];
  const float* b1 = (const float*)d_in[2];
  const float* W2 = (const float*)d_in[3];
  const float* b2 = (const float*)d_in[4];
  const float* W3 = (const float*)d_in[5];
  const float* b3 = (const float*)d_in[6];
  float* out = (float*)d_out;

  // Part 1: all-(b,t) MLP -> out channels [32,48). 131072 rows / 16 per WG.
  mlp_all_kernel<<<dim3((BB * TT) / 16), dim3(512), 0, stream>>>(
      inp, W1, b1, W2, b2, W3, b3, out);

  // Part 2: 511-step rollout -> out channels [0,32). 256 rows / 16 per WG.
  rollout_kernel<<<dim3(BB / 16), dim3(512), 0, stream>>>(
      inp, W1, b1, W2, b2, W3, b3, out);
}